// BPMLLLoss_33139967656428
// MI455X (gfx1250) — compile-verified
//
#include <hip/hip_runtime.h>
#include <math.h>

typedef __attribute__((ext_vector_type(2))) float v2f;
typedef __attribute__((ext_vector_type(8))) float v8f;

#define BROWS 128
#define LCOLS 1024
#define ROW_TILE 16
#define COL_BLKS 4
#define COLS_PER_BLK (LCOLS / COL_BLKS)          // 256
#define WAVES_PER_BLOCK 8
#define COLS_PER_WAVE (COLS_PER_BLK / WAVES_PER_BLOCK)  // 32
#define KSTEPS (COLS_PER_WAVE / 4)               // 8

// Kernel 1: per (16-row tile, 256-col slice) block, compute partial row sums of
//   sp = y*exp(-c), sn = (1-y)*exp(c), yc = y
// using v_wmma_f32_16x16x4_f32 against a ones B-matrix (exact f32 reduction).
__global__ __launch_bounds__(256) void bpmll_rowpart_kernel(
    const float* __restrict__ c, const int* __restrict__ y,
    float* __restrict__ sp_part, float* __restrict__ sn_part,
    float* __restrict__ yc_part)
{
    const int tid    = threadIdx.x;
    const int lane   = tid & 31;
    const int wave   = tid >> 5;
    const int rowblk = blockIdx.x >> 2;   // 0..7
    const int colblk = blockIdx.x & 3;    // 0..3
    const int m      = lane & 15;         // A-matrix row within tile
    const int khalf  = (lane >> 4) * 2;   // lanes 0-15 -> K={0,1}, lanes 16-31 -> K={2,3}

    const int row      = rowblk * ROW_TILE + m;
    const int colbase  = colblk * COLS_PER_BLK + wave * COLS_PER_WAVE + khalf;

    const float* crow = c + (size_t)row * LCOLS;
    const int*   yrow = y + (size_t)row * LCOLS;

    v2f ones = {1.0f, 1.0f};
    v8f acc_sp = {};
    v8f acc_sn = {};
    v8f acc_yc = {};

#pragma unroll
    for (int it = 0; it < KSTEPS; ++it) {
        const int col = colbase + it * 4;
        const float c0 = crow[col];
        const float c1 = crow[col + 1];
        const float y0 = (float)yrow[col];
        const float y1 = (float)yrow[col + 1];

        v2f a_sp = { y0 * expf(-c0),          y1 * expf(-c1)          };
        v2f a_sn = { (1.0f - y0) * expf(c0),  (1.0f - y1) * expf(c1)  };
        v2f a_yc = { y0,                      y1                      };

        // D = A x ones(4x16) + C : every D column = row-sum over this K step.
        acc_sp = __builtin_amdgcn_wmma_f32_16x16x4_f32(
            false, a_sp, false, ones, (short)0, acc_sp, false, false);
        acc_sn = __builtin_amdgcn_wmma_f32_16x16x4_f32(
            false, a_sn, false, ones, (short)0, acc_sn, false, false);
        acc_yc = __builtin_amdgcn_wmma_f32_16x16x4_f32(
            false, a_yc, false, ones, (short)0, acc_yc, false, false);
    }

    // D layout (f32 16x16): VGPR v, lanes 0-15 -> M=v, lanes 16-31 -> M=v+8,
    // value replicated across all 16 N-lanes. Lane 0 and lane 16 carry all rows.
    __shared__ float lsp[WAVES_PER_BLOCK][ROW_TILE];
    __shared__ float lsn[WAVES_PER_BLOCK][ROW_TILE];
    __shared__ float lyc[WAVES_PER_BLOCK][ROW_TILE];

    if ((lane & 15) == 0) {
        const int rbase = (lane >> 4) * 8;
#pragma unroll
        for (int v = 0; v < 8; ++v) {
            lsp[wave][rbase + v] = acc_sp[v];
            lsn[wave][rbase + v] = acc_sn[v];
            lyc[wave][rbase + v] = acc_yc[v];
        }
    }
    __syncthreads();

    if (tid < ROW_TILE) {
        float SP = 0.0f, SN = 0.0f, YC = 0.0f;
#pragma unroll
        for (int w = 0; w < WAVES_PER_BLOCK; ++w) {
            SP += lsp[w][tid];
            SN += lsn[w][tid];
            YC += lyc[w][tid];
        }
        const int r = rowblk * ROW_TILE + tid;
        sp_part[colblk * BROWS + r] = SP;
        sn_part[colblk * BROWS + r] = SN;
        yc_part[colblk * BROWS + r] = YC;
    }
}

// Kernel 2: combine column-slice partials per row, apply the BP-MLL per-row
// formula, deterministic tree-reduce to the mean.
__global__ __launch_bounds__(128) void bpmll_final_kernel(
    const float* __restrict__ sp_part, const float* __restrict__ sn_part,
    const float* __restrict__ yc_part, float* __restrict__ out)
{
    const int r = threadIdx.x;  // 0..127 (one row per thread)
    float SP = 0.0f, SN = 0.0f, YC = 0.0f;
#pragma unroll
    for (int b = 0; b < COL_BLKS; ++b) {
        SP += sp_part[b * BROWS + r];
        SN += sn_part[b * BROWS + r];
        YC += yc_part[b * BROWS + r];
    }
    const float YBAR = (float)LCOLS - YC;
    const float val  = (SP * SN) / (YC * YBAR + 1e-7f);

    __shared__ float s[BROWS];
    s[r] = val;
    __syncthreads();
#pragma unroll
    for (int off = BROWS / 2; off > 0; off >>= 1) {
        if (r < off) s[r] += s[r + off];
        __syncthreads();
    }
    if (r == 0) out[0] = s[0] * (1.0f / (float)BROWS);
}

extern "C" void kernel_launch(void* const* d_in, const int* in_sizes, int n_in,
                              void* d_out, int out_size, void* d_ws, size_t ws_size,
                              hipStream_t stream) {
    (void)in_sizes; (void)n_in; (void)out_size; (void)ws_size;

    const float* c = (const float*)d_in[0];
    const int*   y = (const int*)d_in[1];

    float* ws  = (float*)d_ws;
    float* sp_part = ws;                       // [4][128]
    float* sn_part = ws + COL_BLKS * BROWS;    // [4][128]
    float* yc_part = ws + 2 * COL_BLKS * BROWS;// [4][128]

    bpmll_rowpart_kernel<<<dim3(8 * COL_BLKS), dim3(256), 0, stream>>>(
        c, y, sp_part, sn_part, yc_part);
    bpmll_final_kernel<<<dim3(1), dim3(128), 0, stream>>>(
        sp_part, sn_part, yc_part, (float*)d_out);
}